// ThresholdDecision_86011015069984
// MI455X (gfx1250) — compile-verified
//
#include <hip/hip_runtime.h>

// Problem constants from the reference
#define NB 4096          // batch rows
#define NL 4096          // sequence length
#define NC 3             // channels
#define ROW_FLOATS (NL * NC)      // 12288 floats = 48 KiB per row
#define THREADS 256
#define F4_PER_THREAD (ROW_FLOATS / 4 / THREADS)   // 12 float4 copies per thread

#if defined(__has_builtin)
#  if __has_builtin(__builtin_amdgcn_global_load_async_to_lds_b128)
#    define HAVE_ASYNC_LDS 1
#  endif
#endif

// Builtin signature (from clang diagnostic): pointers to 4xi32 vectors,
// src in addrspace(1) (global), dst in addrspace(3) (LDS).
typedef int v4i __attribute__((ext_vector_type(4)));
typedef __attribute__((address_space(1))) v4i g_v4i;   // global
typedef __attribute__((address_space(3))) v4i l_v4i;   // LDS

__global__ __launch_bounds__(THREADS)
void ThresholdDecision_kernel(const float* __restrict__ x, int* __restrict__ out) {
    const int row = blockIdx.x;
    const int t   = threadIdx.x;

    __shared__ float        lds_row[ROW_FLOATS];   // 48 KiB row staging
    __shared__ unsigned int masks[THREADS];
    __shared__ int          flag;

    const float* rowp = x + (size_t)row * ROW_FLOATS;

#if HAVE_ASYNC_LDS
    // Stream the whole 48 KiB row into LDS via the CDNA5 async DMA path.
    // Each lane copies 16 B per op; 12 async ops in flight per lane (ASYNCcnt).
    #pragma unroll
    for (int k = 0; k < F4_PER_THREAD; ++k) {
        const int f4 = t + THREADS * k;            // coalesced float4 index
        __builtin_amdgcn_global_load_async_to_lds_b128(
            (g_v4i*)(rowp + (size_t)f4 * 4),
            (l_v4i*)(lds_row + f4 * 4),
            /*offset=*/0, /*cpol=*/0);
    }
#  if defined(__has_builtin) && __has_builtin(__builtin_amdgcn_s_wait_asynccnt)
    __builtin_amdgcn_s_wait_asynccnt(0);
#  else
    asm volatile("s_wait_asynccnt 0" ::: "memory");
#  endif
#else
    // Fallback: plain vectorized copy (still correct, just not the async path).
    #pragma unroll
    for (int k = 0; k < F4_PER_THREAD; ++k) {
        const int f4 = t + THREADS * k;
        ((float4*)lds_row)[f4] = ((const float4*)rowp)[f4];
    }
#endif

    if (t == 0) flag = 0;
    __syncthreads();

    // Each thread owns 16 consecutive sequence positions.
    const int p0 = t * 16;
    unsigned int m = 0u;
    #pragma unroll
    for (int i = 0; i < 16; ++i) {
        const float v = lds_row[(p0 + i) * NC + 2];   // channel 2
        m |= (v > 0.5f ? 1u : 0u) << i;
    }
    // Reference windows cover positions [1, L-2] only: clear pos 0 and pos L-1.
    if (t == 0)           m &= ~1u;
    if (t == THREADS - 1) m &= 0x7FFFu;

    masks[t] = m;
    __syncthreads();

    // Append the next thread's low 3 bits so runs straddling the 16-bit
    // boundary are seen; detect any 4-in-a-row starting in our chunk.
    const unsigned int nxt  = (t < THREADS - 1) ? masks[t + 1] : 0u;
    const unsigned int comb = m | (nxt << 16);
    const unsigned int r    = comb & (comb >> 1) & (comb >> 2) & (comb >> 3);
    if (r & 0xFFFFu) atomicOr(&flag, 1);
    __syncthreads();

    if (t == 0) out[row] = flag;   // int32 0/1 per reference
}

extern "C" void kernel_launch(void* const* d_in, const int* in_sizes, int n_in,
                              void* d_out, int out_size, void* d_ws, size_t ws_size,
                              hipStream_t stream) {
    (void)in_sizes; (void)n_in; (void)out_size; (void)d_ws; (void)ws_size;
    const float* x   = (const float*)d_in[0];
    int*         out = (int*)d_out;
    ThresholdDecision_kernel<<<dim3(NB), dim3(THREADS), 0, stream>>>(x, out);
}